// ResBlock_73924977099027
// MI455X (gfx1250) — compile-verified
//
#include <hip/hip_runtime.h>

#define N_NODES 10000
#define E_EDGES 160000
#define C_CH    8
#define B_BATCH 64
#define EPS_LN  1e-5f

typedef __attribute__((ext_vector_type(2))) float v2f;
typedef __attribute__((ext_vector_type(8))) float v8f;

// ---------------------------------------------------------------------------
// 1) Transpose x (B=64, E) -> xT ((E+1), 64); row E is zeroed separately and
//    serves as the sentinel "zero edge" for CSR padding.
// ---------------------------------------------------------------------------
__global__ void transpose_kernel(const float* __restrict__ x,
                                 float* __restrict__ xT) {
    __shared__ float tile[64][65];
    const int e0 = blockIdx.x * 64;
    for (int i = threadIdx.x; i < 64 * 64; i += blockDim.x) {
        int b = i >> 6, el = i & 63;
        tile[b][el] = x[(size_t)b * E_EDGES + e0 + el];
    }
    __syncthreads();
    for (int i = threadIdx.x; i < 64 * 64; i += blockDim.x) {
        int el = i >> 6, b = i & 63;
        xT[(size_t)(e0 + el) * 64 + b] = tile[b][el];
    }
}

__global__ void zero_xt_row_kernel(float* __restrict__ xT) {
    // zero sentinel row E (64 floats)
    xT[(size_t)E_EDGES * B_BATCH + threadIdx.x] = 0.f;
}

// Repack w1 (E,8) -> w1p16 ((E+1),16): channels 8..15 zero, row E zero.
// Removes the c<8 lane guard in the WMMA A-fragment load.
__global__ void pad_w1_kernel(const float* __restrict__ w1,
                              float* __restrict__ w1p16) {
    int i = blockIdx.x * blockDim.x + threadIdx.x;
    if (i >= (E_EDGES + 1) * 16) return;
    int e = i >> 4, c = i & 15;
    w1p16[i] = (e < E_EDGES && c < C_CH) ? w1[e * C_CH + c] : 0.f;
}

// ---------------------------------------------------------------------------
// 2) CSR-by-destination with rows padded to a multiple of 4 (pad slots hold
//    the sentinel edge E). Deterministic: rows are insertion-sorted.
// ---------------------------------------------------------------------------
__global__ void zero_counts_kernel(int* __restrict__ count) {
    int i = blockIdx.x * blockDim.x + threadIdx.x;
    if (i < N_NODES) count[i] = 0;
}

__global__ void count_kernel(const int* __restrict__ dst,
                             int* __restrict__ count) {
    int e = blockIdx.x * blockDim.x + threadIdx.x;
    if (e < E_EDGES) atomicAdd(&count[dst[e]], 1);
}

// Exclusive scan of padded degrees round_up(deg,4).
__global__ void scan_kernel(const int* __restrict__ count,
                            int* __restrict__ row_start,
                            int* __restrict__ cursor, int n) {
    __shared__ int sh[1024];
    __shared__ int carry_sh;
    const int tid = threadIdx.x;
    if (tid == 0) carry_sh = 0;
    __syncthreads();
    for (int base = 0; base < n; base += 1024) {
        int i = base + tid;
        int v = (i < n) ? ((count[i] + 3) & ~3) : 0;  // padded degree
        sh[tid] = v;
        __syncthreads();
        for (int off = 1; off < 1024; off <<= 1) {
            int t = (tid >= off) ? sh[tid - off] : 0;
            __syncthreads();
            sh[tid] += t;
            __syncthreads();
        }
        int carry = carry_sh;
        if (i < n) {
            int excl = sh[tid] + carry - v;
            row_start[i] = excl;
            cursor[i]    = excl;
        }
        __syncthreads();
        if (tid == 1023) carry_sh = carry + sh[1023];
        __syncthreads();
    }
    if (tid == 0) row_start[n] = carry_sh;  // total padded length
}

__global__ void fill_kernel(const int* __restrict__ dst,
                            int* __restrict__ cursor,
                            int* __restrict__ csr) {
    int e = blockIdx.x * blockDim.x + threadIdx.x;
    if (e < E_EDGES) {
        int p = atomicAdd(&cursor[dst[e]], 1);
        csr[p] = e;
    }
}

// Sort each row (determinism) and fill pad slots with sentinel edge E.
__global__ void sort_pad_kernel(const int* __restrict__ row_start,
                                const int* __restrict__ cursor,
                                int* __restrict__ csr) {
    int n = blockIdx.x * blockDim.x + threadIdx.x;
    if (n >= N_NODES) return;
    const int s = row_start[n];
    const int t = cursor[n];          // s + true degree
    const int p = row_start[n + 1];   // padded end
    for (int i = s + 1; i < t; ++i) { // avg degree ~16: insertion sort
        int key = csr[i];
        int j = i - 1;
        while (j >= s && csr[j] > key) { csr[j + 1] = csr[j]; --j; }
        csr[j + 1] = key;
    }
    for (int i = t; i < p; ++i) csr[i] = E_EDGES;  // sentinel zero edge
}

// ---------------------------------------------------------------------------
// 3) Per-node segment-GEMM with V_WMMA_F32_16X16X4_F32 + fused
//    b1/LayerNorm/gamma,beta/ELU epilogue. Branch-free inner loop:
//    all operand rows exist (sentinel rows are zero), rows padded to K%4==0.
//    Two accumulators -> two independent WMMA chains per 8-edge iteration.
// ---------------------------------------------------------------------------
__global__ void __launch_bounds__(128)
accum_norm_kernel(const float* __restrict__ xT,
                  const float* __restrict__ w1p16,
                  const float* __restrict__ b1,
                  const float* __restrict__ gamma,
                  const float* __restrict__ beta,
                  const int*   __restrict__ row_start,
                  const int*   __restrict__ csr,
                  float* __restrict__ h) {
    const int n    = blockIdx.x;
    const int tid  = threadIdx.x;
    const int lane = tid & 31;
    const int wave = tid >> 5;
    const int l16  = lane & 15;
    const int hi   = lane >> 4;          // 0: K slots {0,1}, 1: K slots {2,3}
    const int b0   = wave * 16;
    const int bl   = b0 + l16;

    const int s    = row_start[n];
    const int degp = row_start[n + 1] - s;   // multiple of 4
    const int koff = 2 * hi;

    v8f acc0 = {}, acc1 = {};
    int base = 0;
    for (; base + 8 <= degp; base += 8) {
        // edge-id pairs: (s+base+koff) is even and csr is 256B-aligned -> int2 ok
        const int2 ep0 = *reinterpret_cast<const int2*>(&csr[s + base + koff]);
        const int2 ep1 = *reinterpret_cast<const int2*>(&csr[s + base + 4 + koff]);
        v2f a0, b0v, a1, b1v;
        a0.x  = w1p16[ep0.x * 16 + l16];
        a0.y  = w1p16[ep0.y * 16 + l16];
        b0v.x = xT[ep0.x * B_BATCH + bl];
        b0v.y = xT[ep0.y * B_BATCH + bl];
        a1.x  = w1p16[ep1.x * 16 + l16];
        a1.y  = w1p16[ep1.y * 16 + l16];
        b1v.x = xT[ep1.x * B_BATCH + bl];
        b1v.y = xT[ep1.y * B_BATCH + bl];
        acc0 = __builtin_amdgcn_wmma_f32_16x16x4_f32(
            false, a0, false, b0v, (short)0, acc0, false, false);
        acc1 = __builtin_amdgcn_wmma_f32_16x16x4_f32(
            false, a1, false, b1v, (short)0, acc1, false, false);
    }
    if (base < degp) {  // final K=4 chunk
        const int2 ep0 = *reinterpret_cast<const int2*>(&csr[s + base + koff]);
        v2f a0, b0v;
        a0.x  = w1p16[ep0.x * 16 + l16];
        a0.y  = w1p16[ep0.y * 16 + l16];
        b0v.x = xT[ep0.x * B_BATCH + bl];
        b0v.y = xT[ep0.y * B_BATCH + bl];
        acc0 = __builtin_amdgcn_wmma_f32_16x16x4_f32(
            false, a0, false, b0v, (short)0, acc0, false, false);
    }

    // Fused epilogue: lane l<16 holds channels 0..7 of batch row b0+l.
    float v[C_CH];
    float mu = 0.f;
#pragma unroll
    for (int r = 0; r < C_CH; ++r) {
        v[r] = acc0[r] + acc1[r] + b1[n * C_CH + r];
        mu += v[r];
    }
    mu *= 0.125f;
    float var = 0.f;
#pragma unroll
    for (int r = 0; r < C_CH; ++r) {
        float d = v[r] - mu;
        v[r] = d;
        var += d * d;
    }
    var *= 0.125f;
    const float rs = rsqrtf(var + EPS_LN);
#pragma unroll
    for (int r = 0; r < C_CH; ++r) {
        float t = v[r] * rs * gamma[n * C_CH + r] + beta[n * C_CH + r];
        v[r] = (t > 0.f) ? t : (expf(t) - 1.f);   // ELU(alpha=1)
    }
    if (hi == 0) {
        float* dst = &h[((size_t)n * B_BATCH + bl) * C_CH];
        reinterpret_cast<float4*>(dst)[0] = make_float4(v[0], v[1], v[2], v[3]);
        reinterpret_cast<float4*>(dst)[1] = make_float4(v[4], v[5], v[6], v[7]);
    }
}

// ---------------------------------------------------------------------------
// 4) Gather + dot over C + bias + residual. e-major threads: coalesced x/out;
//    h read is one contiguous 32B (float4 x2) served from L2-resident h.
// ---------------------------------------------------------------------------
__global__ void output_kernel(const float* __restrict__ x,
                              const float* __restrict__ h,
                              const float* __restrict__ w3,
                              const float* __restrict__ b3,
                              const int*   __restrict__ src,
                              float* __restrict__ out) {
    const int e = blockIdx.x * blockDim.x + threadIdx.x;
    const int b = blockIdx.y;
    if (e >= E_EDGES) return;
    const int sn = src[e];
    const float4* hp = reinterpret_cast<const float4*>(
        &h[((size_t)sn * B_BATCH + b) * C_CH]);
    const float4* wp = reinterpret_cast<const float4*>(&w3[(size_t)e * C_CH]);
    const float4 h0 = hp[0], h1 = hp[1];
    const float4 w0 = wp[0], w1v = wp[1];
    float dot = h0.x * w0.x + h0.y * w0.y + h0.z * w0.z + h0.w * w0.w +
                h1.x * w1v.x + h1.y * w1v.y + h1.z * w1v.z + h1.w * w1v.w;
    const size_t idx = (size_t)b * E_EDGES + e;
    out[idx] = dot + b3[e] + x[idx];
}

// ---------------------------------------------------------------------------
extern "C" void kernel_launch(void* const* d_in, const int* in_sizes, int n_in,
                              void* d_out, int out_size, void* d_ws,
                              size_t ws_size, hipStream_t stream) {
    const float* x     = (const float*)d_in[0];  // (64, 160000)
    const float* w1    = (const float*)d_in[1];  // (E, 8)
    const float* b1    = (const float*)d_in[2];  // (N, 8)
    const float* gamma = (const float*)d_in[3];  // (N, 8)
    const float* beta  = (const float*)d_in[4];  // (N, 8)
    const float* w3    = (const float*)d_in[5];  // (E, 8)
    const float* b3    = (const float*)d_in[6];  // (E,)
    const int*   esrc  = (const int*)d_in[7];    // (E,)
    const int*   edst  = (const int*)d_in[8];    // (E,)
    float* out = (float*)d_out;                  // (64, 160000)

    // Workspace carve-out (256B aligned): ~73.5 MB total.
    char* ws = (char*)d_ws;
    size_t off = 0;
    auto carve = [&](size_t bytes) -> void* {
        void* p = ws + off;
        off = (off + bytes + 255) & ~(size_t)255;
        return p;
    };
    float* xT        = (float*)carve((size_t)(E_EDGES + 1) * B_BATCH * 4);
    float* w1p16     = (float*)carve((size_t)(E_EDGES + 1) * 16 * 4);
    float* h         = (float*)carve((size_t)N_NODES * B_BATCH * C_CH * 4);
    int*   count     = (int*)carve((size_t)N_NODES * 4);
    int*   row_start = (int*)carve((size_t)(N_NODES + 1) * 4);
    int*   cursor    = (int*)carve((size_t)N_NODES * 4);
    int*   csr       = (int*)carve((size_t)(E_EDGES + 4 * N_NODES) * 4);
    (void)ws_size; (void)in_sizes; (void)n_in; (void)out_size;

    transpose_kernel<<<E_EDGES / 64, 256, 0, stream>>>(x, xT);
    zero_xt_row_kernel<<<1, 64, 0, stream>>>(xT);
    pad_w1_kernel<<<((E_EDGES + 1) * 16 + 255) / 256, 256, 0, stream>>>(w1, w1p16);
    zero_counts_kernel<<<(N_NODES + 255) / 256, 256, 0, stream>>>(count);
    count_kernel<<<(E_EDGES + 255) / 256, 256, 0, stream>>>(edst, count);
    scan_kernel<<<1, 1024, 0, stream>>>(count, row_start, cursor, N_NODES);
    fill_kernel<<<(E_EDGES + 255) / 256, 256, 0, stream>>>(edst, cursor, csr);
    sort_pad_kernel<<<(N_NODES + 255) / 256, 256, 0, stream>>>(row_start, cursor, csr);
    accum_norm_kernel<<<N_NODES, 128, 0, stream>>>(xT, w1p16, b1, gamma, beta,
                                                   row_start, csr, h);
    dim3 og((E_EDGES + 255) / 256, B_BATCH);
    output_kernel<<<og, 256, 0, stream>>>(x, h, w3, b3, esrc, out);
}